// topk_crossEntrophy_54889682043506
// MI455X (gfx1250) — compile-verified
//
#include <hip/hip_runtime.h>
#include <hip/hip_bf16.h>
#include <float.h>

// ---------------------------------------------------------------------------
// topk cross-entropy:
//   loss_i = logsumexp(input[i,:]) - input[i, target[i]]   (i = 0..N-1)
//   out    = mean(top_k(loss, k)),  k = int(0.7*N)
//
// Kernel 1: one block per row, single-pass online softmax (matrix read once:
//           1.65 GB @ 23.3 TB/s => ~71us floor). Rescale exp only when the
//           running max actually moves (steady state: 1 exp per element).
// Kernel 2: one block: bitonic sort of N=8192 losses in LDS (descending),
//           zero-pad past k, then exact fp32 top-k sum via
//           V_WMMA_F32_16X16X4_F32 with an all-ones B (D = A*1 + C folds
//           64 values per issue; unguarded LDS loads, wave-uniform bounds).
// ---------------------------------------------------------------------------

typedef float v2f __attribute__((ext_vector_type(2)));
typedef float v8f __attribute__((ext_vector_type(8)));

#define ROW_THREADS 256
#define ROW_WAVES   (ROW_THREADS / 32)
#define NPOW        8192          // power-of-two sort size (N == 8192)
#define SORT_THREADS 1024
#define SORT_WAVES  (SORT_THREADS / 32)

// Merge two online-softmax partial states (m0,s0) <- (m0,s0) + (m1,s1).
__device__ __forceinline__ void merge_ms(float& m0, float& s0, float m1, float s1) {
    float nm = fmaxf(m0, m1);
    // m==-FLT_MAX identities: exp(very negative) -> 0, never NaN.
    s0 = s0 * __expf(m0 - nm) + s1 * __expf(m1 - nm);
    m0 = nm;
}

__global__ void row_loss_kernel(const float* __restrict__ inp,
                                const long long* __restrict__ tgt,
                                float* __restrict__ loss,
                                int C) {
    const int row = blockIdx.x;
    const float* __restrict__ x = inp + (size_t)row * (size_t)C;
    const int tid = threadIdx.x;

    float m = -FLT_MAX;
    float s = 0.0f;

    // --- vectorized main body: float4 streaming loads -----------------------
    const int nvec = C >> 2;                       // # of float4 chunks
    const float4* __restrict__ x4 = (const float4*)x;
    for (int i = tid; i < nvec; i += ROW_THREADS) {
        // stay ~32KB ahead of the load stream (speculative, OOB is dropped)
        __builtin_prefetch(&x4[i + 8 * ROW_THREADS], 0, 1);
        float4 v = x4[i];
        float vm = fmaxf(fmaxf(v.x, v.y), fmaxf(v.z, v.w));
        if (vm > m) {                  // rare after warmup: rescale running sum
            s *= __expf(m - vm);
            m = vm;
        }
        s += __expf(v.x - m) + __expf(v.y - m) +
             __expf(v.z - m) + __expf(v.w - m);
    }
    // --- scalar tail --------------------------------------------------------
    for (int i = (nvec << 2) + tid; i < C; i += ROW_THREADS) {
        float v = x[i];
        if (v > m) {
            s *= __expf(m - v);
            m = v;
        }
        s += __expf(v - m);
    }

    // --- wave32 reduction ---------------------------------------------------
    for (int off = 16; off > 0; off >>= 1) {
        float om = __shfl_xor(m, off, 32);
        float os = __shfl_xor(s, off, 32);
        merge_ms(m, s, om, os);
    }

    // --- cross-wave reduction via LDS (8 waves) -----------------------------
    __shared__ float sm[ROW_WAVES];
    __shared__ float ss[ROW_WAVES];
    const int lane = tid & 31;
    const int wave = tid >> 5;
    if (lane == 0) { sm[wave] = m; ss[wave] = s; }
    __syncthreads();

    if (wave == 0) {
        m = (lane < ROW_WAVES) ? sm[lane] : -FLT_MAX;
        s = (lane < ROW_WAVES) ? ss[lane] : 0.0f;
        for (int off = 16; off > 0; off >>= 1) {
            float om = __shfl_xor(m, off, 32);
            float os = __shfl_xor(s, off, 32);
            merge_ms(m, s, om, os);
        }
        if (lane == 0) {
            long long t = tgt[row];
            float xt = x[(int)t];
            loss[row] = (m + __logf(s)) - xt;
        }
    }
}

__global__ void topk_mean_kernel(const float* __restrict__ loss,
                                 float* __restrict__ out,
                                 int n, int k) {
    __shared__ float sd[NPOW];
    __shared__ float wsum[SORT_WAVES];
    const int tid  = threadIdx.x;
    const int lane = tid & 31;
    const int wave = tid >> 5;

    // load losses, pad to power-of-two with -FLT_MAX
    for (int i = tid; i < NPOW; i += SORT_THREADS)
        sd[i] = (i < n) ? loss[i] : -FLT_MAX;

    // ---- bitonic sort, descending -----------------------------------------
    for (int sz = 2; sz <= NPOW; sz <<= 1) {
        for (int stride = sz >> 1; stride > 0; stride >>= 1) {
            __syncthreads();
            for (int i = tid; i < NPOW; i += SORT_THREADS) {
                int j = i ^ stride;
                if (j > i) {
                    bool desc = ((i & sz) == 0);      // descending overall
                    float a = sd[i], b = sd[j];
                    bool swap = desc ? (a < b) : (a > b);
                    if (swap) { sd[i] = b; sd[j] = a; }
                }
            }
        }
    }
    __syncthreads();

    // zero-pad past k so the WMMA loop can load unguarded (0 = additive id.)
    for (int i = k + tid; i < NPOW; i += SORT_THREADS)
        sd[i] = 0.0f;
    __syncthreads();

    // ---- exact fp32 top-k sum on the matrix pipe ---------------------------
    // Each wave consumes 64-value chunks of sd[0..k) as a 16x4 fp32 A tile;
    // B = ones(4x16), so D = A*B + C accumulates each chunk's values into the
    // row-sums of C (every C column identical). Wave-uniform loop bounds keep
    // EXEC all-1s at every WMMA (ISA 7.12 requirement); tail chunk reads the
    // zeroed pad region, never past NPOW (last base = 5696, reads < 5760).
    v8f c = {};
    v2f b; b.x = 1.0f; b.y = 1.0f;
    for (int base = wave * 64; base < k; base += SORT_WAVES * 64) {
        v2f a;
        a.x = sd[base + lane];
        a.y = sd[base + 32 + lane];
        c = __builtin_amdgcn_wmma_f32_16x16x4_f32(
                /*neg_a=*/false, a, /*neg_b=*/false, b,
                /*c_mod=*/(short)0, c, /*reuse_a=*/false, /*reuse_b=*/false);
    }
    // Full C tile holds each row-sum replicated across 16 columns:
    // sum(all 256 C elems) = 16 * (chunk total). Reduce and scale by 1/16.
    float p = c[0] + c[1] + c[2] + c[3] + c[4] + c[5] + c[6] + c[7];
    for (int off = 16; off > 0; off >>= 1)
        p += __shfl_xor(p, off, 32);
    if (lane == 0) wsum[wave] = p * (1.0f / 16.0f);
    __syncthreads();

    if (tid == 0) {
        float tot = 0.0f;
        for (int w = 0; w < SORT_WAVES; ++w) tot += wsum[w];
        out[0] = tot / (float)k;
    }
}

extern "C" void kernel_launch(void* const* d_in, const int* in_sizes, int n_in,
                              void* d_out, int out_size, void* d_ws, size_t ws_size,
                              hipStream_t stream) {
    const float*     inp = (const float*)d_in[0];
    const long long* tgt = (const long long*)d_in[1];   // int64 targets
    float* out = (float*)d_out;

    const int total = in_sizes[0];      // N * C
    const int n     = in_sizes[1];      // N  (8192)
    const int C     = total / n;        // C  (50257)
    const int k     = (int)(0.7 * (double)n);   // matches int(TOP_K * n)

    float* losses = (float*)d_ws;       // N floats of scratch

    row_loss_kernel<<<n, ROW_THREADS, 0, stream>>>(inp, tgt, losses, C);
    topk_mean_kernel<<<1, SORT_THREADS, 0, stream>>>(losses, out, n, k);
}